// HGT_57758720196946
// MI455X (gfx1250) — compile-verified
//
#include <hip/hip_runtime.h>
#include <cstddef>
#include <cstdint>

// ---------------------------------------------------------------------------
// HGT forward for MI455X (gfx1250). All dense GEMMs + attention use
// V_WMMA_F32_16X16X4_F32 (fp32-exact matrix path, wave32).
// ---------------------------------------------------------------------------

typedef float v2f __attribute__((ext_vector_type(2)));
typedef float v8f __attribute__((ext_vector_type(8)));

__device__ __forceinline__ v8f wmma4f32(v2f a, v2f b, v8f c) {
  // (neg_a, A, neg_b, B, c_mod, C, reuse_a, reuse_b) -> v_wmma_f32_16x16x4_f32
  return __builtin_amdgcn_wmma_f32_16x16x4_f32(false, a, false, b, (short)0, c,
                                               false, false);
}

__device__ __forceinline__ float dev_sigmoid(float x) {
  return 1.0f / (1.0f + __expf(-x));
}
__device__ __forceinline__ float dev_gelu(float x) {
  return 0.5f * x * (1.0f + tanhf(0.7978845608028654f * (x + 0.044715f * x * x * x)));
}

// C[M x Nc] = act( (accum ? C : 0) + A[M x K] @ B[K x Nc] + bias )
// grid = (Nc/64, M/16), block = 32 (one wave). Nc % 64 == 0, K % 4 == 0.
__global__ __launch_bounds__(32) void k_gemm(const float* __restrict__ A,
                                             const float* __restrict__ B,
                                             const float* __restrict__ bias,
                                             float* __restrict__ C, int K,
                                             int lda, int ldb, int ldc,
                                             int accum, int act) {
  const int lane = threadIdx.x;
  const int m0 = blockIdx.y << 4;
  const int n0 = blockIdx.x << 6;
  const int mr = lane & 15;
  const int hi = lane >> 4;
  const int kh = hi << 1;

  v8f zero = {0.f, 0.f, 0.f, 0.f, 0.f, 0.f, 0.f, 0.f};
  v8f acc[4];
  if (accum) {
#pragma unroll
    for (int nt = 0; nt < 4; ++nt)
#pragma unroll
      for (int r = 0; r < 8; ++r)
        acc[nt][r] = C[(size_t)(m0 + r + hi * 8) * ldc + n0 + nt * 16 + mr];
  } else {
#pragma unroll
    for (int nt = 0; nt < 4; ++nt) acc[nt] = zero;
  }

  const float* Arow = A + (size_t)(m0 + mr) * lda;
  for (int k0 = 0; k0 < K; k0 += 4) {
    v2f a;
    a.x = Arow[k0 + kh];
    a.y = Arow[k0 + kh + 1];
    const float* Bp = B + (size_t)(k0 + kh) * ldb + n0 + mr;
#pragma unroll
    for (int nt = 0; nt < 4; ++nt) {
      v2f b;
      b.x = Bp[nt * 16];
      b.y = Bp[ldb + nt * 16];
      acc[nt] = wmma4f32(a, b, acc[nt]);
    }
  }

#pragma unroll
  for (int nt = 0; nt < 4; ++nt)
#pragma unroll
    for (int r = 0; r < 8; ++r) {
      float v = acc[nt][r];
      int col = n0 + nt * 16 + mr;
      if (bias) v += bias[col];
      if (act == 1) v = fmaxf(v, 0.0f);
      else if (act == 2) v = dev_gelu(v);
      C[(size_t)(m0 + r + hi * 8) * ldc + col] = v;
    }
}

// Streaming (flash) softmax attention, d = 128. One wave handles 16 query
// rows; online softmax in WMMA C-layout; P goes through LDS to get A-layout.
// O[q, :] (+)= softmax(Q K^T / sqrt(128)) @ V
__global__ __launch_bounds__(32) void k_attn(const float* __restrict__ Q,
                                             const float* __restrict__ K,
                                             const float* __restrict__ V,
                                             float* __restrict__ O, int n,
                                             int ldo, int addout) {
  __shared__ float pbuf[256];
  const int lane = threadIdx.x;
  const int q0 = blockIdx.x << 4;
  const int mr = lane & 15;
  const int hi = lane >> 4;
  const int kh = hi << 1;
  const float scale = 0.08838834764831845f;  // 1/sqrt(128)

  v2f Qa[32];
  const float* Qrow = Q + (size_t)(q0 + mr) * 128;
#pragma unroll
  for (int c = 0; c < 32; ++c) {
    Qa[c].x = Qrow[4 * c + kh];
    Qa[c].y = Qrow[4 * c + kh + 1];
  }

  v8f zero = {0.f, 0.f, 0.f, 0.f, 0.f, 0.f, 0.f, 0.f};
  v8f Ov[8];
#pragma unroll
  for (int t = 0; t < 8; ++t) Ov[t] = zero;
  float mrow[8], lrow[8];
#pragma unroll
  for (int r = 0; r < 8; ++r) {
    mrow[r] = -3.0e38f;
    lrow[r] = 0.f;
  }

  for (int t0 = 0; t0 < n; t0 += 16) {
    v8f s = zero;
    const float* Krow = K + (size_t)(t0 + mr) * 128;
    for (int c = 0; c < 32; ++c) {
      v2f b;
      b.x = Krow[4 * c + kh];
      b.y = Krow[4 * c + kh + 1];
      s = wmma4f32(Qa[c], b, s);
    }
    float p[8];
#pragma unroll
    for (int r = 0; r < 8; ++r) {
      float x = s[r] * scale;
      float tm = x;
#pragma unroll
      for (int off = 1; off < 16; off <<= 1) tm = fmaxf(tm, __shfl_xor(tm, off, 32));
      float nm = fmaxf(mrow[r], tm);
      float sc = __expf(mrow[r] - nm);
      float pe = __expf(x - nm);
      float su = pe;
#pragma unroll
      for (int off = 1; off < 16; off <<= 1) su += __shfl_xor(su, off, 32);
      lrow[r] = lrow[r] * sc + su;
      mrow[r] = nm;
      p[r] = pe;
#pragma unroll
      for (int nt = 0; nt < 8; ++nt) Ov[nt][r] *= sc;
    }
#pragma unroll
    for (int r = 0; r < 8; ++r) pbuf[(r + 8 * hi) * 16 + mr] = p[r];
    __syncthreads();
    v2f pa[4];
#pragma unroll
    for (int cc = 0; cc < 4; ++cc) {
      pa[cc].x = pbuf[mr * 16 + 4 * cc + kh];
      pa[cc].y = pbuf[mr * 16 + 4 * cc + kh + 1];
    }
#pragma unroll
    for (int nt = 0; nt < 8; ++nt) {
#pragma unroll
      for (int cc = 0; cc < 4; ++cc) {
        v2f vb;
        const float* Vp = V + (size_t)(t0 + 4 * cc + kh) * 128 + nt * 16 + mr;
        vb.x = Vp[0];
        vb.y = Vp[128];
        Ov[nt] = wmma4f32(pa[cc], vb, Ov[nt]);
      }
    }
    __syncthreads();
  }
#pragma unroll
  for (int nt = 0; nt < 8; ++nt)
#pragma unroll
    for (int r = 0; r < 8; ++r) {
      size_t idx = (size_t)(q0 + r + 8 * hi) * ldo + nt * 16 + mr;
      float v = Ov[nt][r] / lrow[r];
      if (addout) O[idx] += v;
      else O[idx] = v;
    }
}

// ------------------------- elementwise / edge kernels ----------------------

__global__ void k_fill_f32(float* p, float v, size_t n) {
  for (size_t i = blockIdx.x * (size_t)blockDim.x + threadIdx.x; i < n;
       i += (size_t)gridDim.x * blockDim.x)
    p[i] = v;
}
__global__ void k_fill_i32(int* p, int v, size_t n) {
  for (size_t i = blockIdx.x * (size_t)blockDim.x + threadIdx.x; i < n;
       i += (size_t)gridDim.x * blockDim.x)
    p[i] = v;
}
__global__ void k_gelu_inplace(float* p, size_t n) {
  for (size_t i = blockIdx.x * (size_t)blockDim.x + threadIdx.x; i < n;
       i += (size_t)gridDim.x * blockDim.x)
    p[i] = dev_gelu(p[i]);
}
__global__ void k_skip_combine(float* x, const float* o, const float* skip,
                               size_t n) {
  float beta = dev_sigmoid(skip[0]);
  for (size_t i = blockIdx.x * (size_t)blockDim.x + threadIdx.x; i < n;
       i += (size_t)gridDim.x * blockDim.x)
    x[i] = beta * o[i] + (1.0f - beta) * x[i];
}
__global__ void k_copy_cols(const float* src, int lds_, float* dst, int ldd,
                            int rows, int cols) {
  size_t i = blockIdx.x * (size_t)blockDim.x + threadIdx.x;
  if (i >= (size_t)rows * cols) return;
  int r = (int)(i / cols), c = (int)(i % cols);
  dst[(size_t)r * ldd + c] = src[(size_t)r * lds_ + c];
}

// logit[e,h] = (sum_f q[dst,h,f] * sum_d k[src,h,d]*a[h,d,f]) * p_rel[h]/4
__global__ void k_edge_logit(const float* __restrict__ k,
                             const float* __restrict__ q,
                             const float* __restrict__ arel,
                             const float* __restrict__ prel,
                             const int* __restrict__ src,
                             const int* __restrict__ dst, int E, float* logit,
                             int* mmax) {
  int i = blockIdx.x * blockDim.x + threadIdx.x;
  if (i >= E * 8) return;
  int e = i >> 3, h = i & 7;
  int sN = src[e], dN = dst[e];
  const float* kv = k + (size_t)sN * 128 + h * 16;
  const float* qv = q + (size_t)dN * 128 + h * 16;
  const float* a = arel + h * 256;
  float acc = 0.f;
  for (int f = 0; f < 16; ++f) {
    float t = 0.f;
    for (int d = 0; d < 16; ++d) t += kv[d] * a[d * 16 + f];
    acc += qv[f] * t;
  }
  acc *= prel[h] * 0.25f;  // / sqrt(HD=16)
  logit[i] = acc;
  int iv = __float_as_int(acc);
  int key = (iv >= 0) ? iv : (iv ^ 0x7fffffff);  // order-preserving map
  atomicMax(&mmax[dN * 8 + h], key);
}

__global__ void k_edge_exp(float* logit, const int* __restrict__ dst,
                           const int* __restrict__ mmax, float* ssum, int E) {
  int i = blockIdx.x * blockDim.x + threadIdx.x;
  if (i >= E * 8) return;
  int e = i >> 3, h = i & 7;
  int key = mmax[dst[e] * 8 + h];
  float m = (key >= 0) ? __int_as_float(key) : __int_as_float(key ^ 0x7fffffff);
  float ex = __expf(logit[i] - m);
  logit[i] = ex;
  atomicAdd(&ssum[dst[e] * 8 + h], ex);
}

// agg[dst,h,f] += alpha[e,h] * sum_d v[src,h,d]*m_rel[h,d,f]
__global__ void k_edge_agg(const float* __restrict__ v,
                           const float* __restrict__ mrel,
                           const float* __restrict__ ex,
                           const int* __restrict__ src,
                           const int* __restrict__ dst,
                           const float* __restrict__ ssum, float* agg, int E) {
  int i = blockIdx.x * blockDim.x + threadIdx.x;
  if (i >= E * 128) return;
  int e = i >> 7, hf = i & 127, h = hf >> 4, f = hf & 15;
  int sN = src[e], dN = dst[e];
  float alpha = ex[e * 8 + h] / (ssum[dN * 8 + h] + 1e-9f);
  const float* vv = v + (size_t)sN * 128 + h * 16;
  const float* mm = mrel + h * 256 + f;
  float ve = 0.f;
  for (int d = 0; d < 16; ++d) ve += vv[d] * mm[d * 16];
  atomicAdd(&agg[(size_t)dN * 128 + hf], alpha * ve);
}

// GRU gates (r,z,n order), writes h' into xt[:, 0:128] (ldc = 256)
__global__ void k_gru_gate(const float* __restrict__ gi,
                           const float* __restrict__ gh,
                           const float* __restrict__ h0, float* xt, int n) {
  int i = blockIdx.x * blockDim.x + threadIdx.x;
  if (i >= n * 128) return;
  int row = i >> 7, j = i & 127;
  const float* gir = gi + (size_t)row * 384;
  const float* ghr = gh + (size_t)row * 384;
  float r = dev_sigmoid(gir[j] + ghr[j]);
  float z = dev_sigmoid(gir[128 + j] + ghr[128 + j]);
  float nn = tanhf(gir[256 + j] + r * ghr[256 + j]);
  float h = h0[(size_t)row * 128 + j];
  xt[(size_t)row * 256 + j] = (1.0f - z) * nn + z * h;
}

// y[row] = dot(h[row,:], w) + b   (final 1-wide heads)
__global__ __launch_bounds__(32) void k_rowdot(const float* __restrict__ h,
                                               const float* __restrict__ w,
                                               const float* __restrict__ b,
                                               float* y, int K) {
  int row = blockIdx.x, lane = threadIdx.x;
  const float* hr = h + (size_t)row * K;
  float s = 0.f;
  for (int k = lane; k < K; k += 32) s += hr[k] * w[k];
#pragma unroll
  for (int off = 16; off; off >>= 1) s += __shfl_xor(s, off, 32);
  if (lane == 0) y[row] = s + b[0];
}

// ------------------------------ host driver --------------------------------

// params pytree leaves (jax flatten: dict keys sorted at every level)
enum {
  P_ATT_S_KW = 0, P_ATT_S_KB, P_ATT_S_QW, P_ATT_S_QB, P_ATT_S_VW, P_ATT_S_VB,
  P_ATT_SU_KW, P_ATT_SU_KB, P_ATT_SU_OW, P_ATT_SU_OB, P_ATT_SU_QW, P_ATT_SU_QB,
  P_ATT_SU_VW, P_ATT_SU_VB,
  P_ATT_W_KW, P_ATT_W_KB, P_ATT_W_QW, P_ATT_W_QB, P_ATT_W_VW, P_ATT_W_VB,
  P_ATT_WU_KW, P_ATT_WU_KB, P_ATT_WU_OW, P_ATT_WU_OB, P_ATT_WU_QW, P_ATT_WU_QB,
  P_ATT_WU_VW, P_ATT_WU_VB,
  P_BACK_N1W, P_BACK_N1B, P_BACK_N2W, P_BACK_N2B,
  P_CONV0 = 32,  // 24 leaves per conv layer
  P_CONV1 = 56,
  P_FCC_W = 80, P_FCC_B, P_FCD_W, P_FCD_B,
  P_GRU_N1_BHH = 84, P_GRU_N1_BIH, P_GRU_N1_WHH, P_GRU_N1_WIH,
  P_GRU_N2_BHH, P_GRU_N2_BIH, P_GRU_N2_WHH, P_GRU_N2_WIH,
  P_GRUI_N1W = 92, P_GRUI_N1B, P_GRUI_N2W, P_GRUI_N2B,
  P_GRUL_N1W = 96, P_GRUL_N1B, P_GRUL_N2W, P_GRUL_N2B,
  P_LIN_N1W = 100, P_LIN_N1B, P_LIN_N2W, P_LIN_N2B,
  P_OUT_N1W = 104, P_OUT_N1B, P_OUT_N2W, P_OUT_N2B, P_OUT_PKW, P_OUT_PKB,
  P_OH_N1W = 110, P_OH_N1B, P_OH_N2W, P_OH_N2B, P_OH_PKW, P_OH_PKB,
  P_NUM = 116
};
// conv layer sub-offsets: a_lin(W,b x2)=0..3, a_rel r12=4,r21=5, k=6..9,
// m_rel=10..11, p_rel=12..13, q=14..17, skip=18..19, v=20..23
enum { C_ALW1 = 0, C_ALB1, C_ALW2, C_ALB2, C_AR12, C_AR21, C_KW1, C_KB1,
       C_KW2, C_KB2, C_MR12, C_MR21, C_PR12, C_PR21, C_QW1, C_QB1, C_QW2,
       C_QB2, C_SK1, C_SK2, C_VW1, C_VB1, C_VW2, C_VB2 };

extern "C" void kernel_launch(void* const* d_in, const int* in_sizes, int n_in,
                              void* d_out, int out_size, void* d_ws,
                              size_t ws_size, hipStream_t stream) {
  (void)n_in; (void)out_size; (void)ws_size;
  const int NNn = 4096;
  const size_t NS = (size_t)NNn;

  const float* x1 = (const float*)d_in[0];
  const float* x2 = (const float*)d_in[1];
  const int* e12 = (const int*)d_in[2];
  const int* e21 = (const int*)d_in[3];
  const float* t1 = (const float*)d_in[4];
  const float* t2 = (const float*)d_in[5];
  const float* b1 = (const float*)d_in[6];
  const float* b2 = (const float*)d_in[7];
  const float* mask_in1 = (const float*)d_in[8];
  const float* mask_in2 = (const float*)d_in[9];
  const float* mask_pack2 = (const float*)d_in[10];
  const float* cmask = (const float*)d_in[11];
  const float* dmask = (const float*)d_in[12];
  const int E = in_sizes[2] / 2;

  const float* P[P_NUM];
  for (int i = 0; i < P_NUM; ++i) P[i] = (const float*)d_in[13 + i];

  float* out = (float*)d_out;

  // ---- workspace arena (floats) ----
  float* W = (float*)d_ws;
  float* xs1 = W;
  float* xs2 = xs1 + NS * 128;
  float* xt1 = xs2 + NS * 128;
  float* xt2 = xt1 + NS * 256;
  float* xall1 = xt2 + NS * 256;
  float* xall2 = xall1 + NS * 512;
  float* xout1 = xall2 + NS * 512;
  float* xo3cat = xout1 + NS * 1024;
  float* xo3pack = xo3cat + NS * 1024;
  float* xout3 = xo3pack + NS * 1024;
  float* scr = xout3 + NS * 1024;  // reused scratch, NS*2048 floats

  auto GEMM = [&](const float* A, const float* B, const float* bias, float* C,
                  int M, int Nc, int K, int lda, int ldb, int ldc, int accum,
                  int act) {
    dim3 g(Nc / 64, M / 16);
    k_gemm<<<g, 32, 0, stream>>>(A, B, bias, C, K, lda, ldb, ldc, accum, act);
  };
  auto EW = [&](size_t n) { return dim3((unsigned)((n + 255) / 256)); };

  // ---- 1. lin_in + relu ----
  GEMM(x1, P[P_LIN_N1W], P[P_LIN_N1B], xs1, NNn, 128, 64, 64, 128, 128, 0, 1);
  GEMM(x2, P[P_LIN_N2W], P[P_LIN_N2B], xs2, NNn, 128, 64, 64, 128, 128, 0, 1);

  // ---- 2. HGT conv x2 ----
  float* kN1 = scr;
  float* kN2 = kN1 + NS * 128;
  float* qN1 = kN2 + NS * 128;
  float* qN2 = qN1 + NS * 128;
  float* vN1 = qN2 + NS * 128;
  float* vN2 = vN1 + NS * 128;
  float* agg1 = vN2 + NS * 128;
  float* agg2 = agg1 + NS * 128;
  int* mb1 = (int*)(agg2 + NS * 128);
  float* sb1 = (float*)mb1 + NS * 8;
  int* mb2 = (int*)(sb1 + NS * 8);
  float* sb2 = (float*)mb2 + NS * 8;
  float* elog = sb2 + NS * 8;       // E*8
  float* otmp = elog + (size_t)E * 8;  // NS*128

  for (int l = 0; l < 2; ++l) {
    const float* const* C = P + (32 + 24 * l);
    GEMM(xs1, C[C_KW1], C[C_KB1], kN1, NNn, 128, 128, 128, 128, 128, 0, 0);
    GEMM(xs2, C[C_KW2], C[C_KB2], kN2, NNn, 128, 128, 128, 128, 128, 0, 0);
    GEMM(xs1, C[C_QW1], C[C_QB1], qN1, NNn, 128, 128, 128, 128, 128, 0, 0);
    GEMM(xs2, C[C_QW2], C[C_QB2], qN2, NNn, 128, 128, 128, 128, 128, 0, 0);
    GEMM(xs1, C[C_VW1], C[C_VB1], vN1, NNn, 128, 128, 128, 128, 128, 0, 0);
    GEMM(xs2, C[C_VW2], C[C_VB2], vN2, NNn, 128, 128, 128, 128, 128, 0, 0);
    k_fill_f32<<<EW(NS * 128), 256, 0, stream>>>(agg1, 0.f, NS * 128);
    k_fill_f32<<<EW(NS * 128), 256, 0, stream>>>(agg2, 0.f, NS * 128);
    // r12: Node1 -> Node2
    k_fill_i32<<<EW(NS * 8), 256, 0, stream>>>(mb2, (int)0x80000000, NS * 8);
    k_fill_f32<<<EW(NS * 8), 256, 0, stream>>>(sb2, 0.f, NS * 8);
    k_edge_logit<<<EW((size_t)E * 8), 256, 0, stream>>>(
        kN1, qN2, C[C_AR12], C[C_PR12], e12, e12 + E, E, elog, mb2);
    k_edge_exp<<<EW((size_t)E * 8), 256, 0, stream>>>(elog, e12 + E, mb2, sb2, E);
    k_edge_agg<<<EW((size_t)E * 128), 256, 0, stream>>>(
        vN1, C[C_MR12], elog, e12, e12 + E, sb2, agg2, E);
    // r21: Node2 -> Node1
    k_fill_i32<<<EW(NS * 8), 256, 0, stream>>>(mb1, (int)0x80000000, NS * 8);
    k_fill_f32<<<EW(NS * 8), 256, 0, stream>>>(sb1, 0.f, NS * 8);
    k_edge_logit<<<EW((size_t)E * 8), 256, 0, stream>>>(
        kN2, qN1, C[C_AR21], C[C_PR21], e21, e21 + E, E, elog, mb1);
    k_edge_exp<<<EW((size_t)E * 8), 256, 0, stream>>>(elog, e21 + E, mb1, sb1, E);
    k_edge_agg<<<EW((size_t)E * 128), 256, 0, stream>>>(
        vN2, C[C_MR21], elog, e21, e21 + E, sb1, agg1, E);
    // a_lin(gelu(agg)) + skip mix
    k_gelu_inplace<<<EW(NS * 128), 256, 0, stream>>>(agg1, NS * 128);
    k_gelu_inplace<<<EW(NS * 128), 256, 0, stream>>>(agg2, NS * 128);
    GEMM(agg1, C[C_ALW1], C[C_ALB1], otmp, NNn, 128, 128, 128, 128, 128, 0, 0);
    k_skip_combine<<<EW(NS * 128), 256, 0, stream>>>(xs1, otmp, C[C_SK1], NS * 128);
    GEMM(agg2, C[C_ALW2], C[C_ALB2], otmp, NNn, 128, 128, 128, 128, 128, 0, 0);
    k_skip_combine<<<EW(NS * 128), 256, 0, stream>>>(xs2, otmp, C[C_SK2], NS * 128);
  }

  // ---- 3. GRU + back_lin -> xt (N x 256) ----
  {
    float* gin = scr;
    float* h0 = gin + NS * 128;
    float* gi = h0 + NS * 128;      // N x 384
    float* gh = gi + NS * 384;      // N x 384
    // Node1
    GEMM(t1, P[P_GRUL_N1W], P[P_GRUL_N1B], gin, NNn, 128, 16, 16, 128, 128, 0, 0);
    GEMM(xs1, P[P_GRUI_N1W], P[P_GRUI_N1B], h0, NNn, 128, 128, 128, 128, 128, 0, 0);
    GEMM(gin, P[P_GRU_N1_WIH], P[P_GRU_N1_BIH], gi, NNn, 384, 128, 128, 384, 384, 0, 0);
    GEMM(h0, P[P_GRU_N1_WHH], P[P_GRU_N1_BHH], gh, NNn, 384, 128, 128, 384, 384, 0, 0);
    k_gru_gate<<<EW(NS * 128), 256, 0, stream>>>(gi, gh, h0, xt1, NNn);
    GEMM(b1, P[P_BACK_N1W], P[P_BACK_N1B], xt1 + 128, NNn, 128, 16, 16, 128, 256, 0, 0);
    // Node2
    GEMM(t2, P[P_GRUL_N2W], P[P_GRUL_N2B], gin, NNn, 128, 16, 16, 128, 128, 0, 0);
    GEMM(xs2, P[P_GRUI_N2W], P[P_GRUI_N2B], h0, NNn, 128, 128, 128, 128, 128, 0, 0);
    GEMM(gin, P[P_GRU_N2_WIH], P[P_GRU_N2_BIH], gi, NNn, 384, 128, 128, 384, 384, 0, 0);
    GEMM(h0, P[P_GRU_N2_WHH], P[P_GRU_N2_BHH], gh, NNn, 384, 128, 128, 384, 384, 0, 0);
    k_gru_gate<<<EW(NS * 128), 256, 0, stream>>>(gi, gh, h0, xt2, NNn);
    GEMM(b2, P[P_BACK_N2W], P[P_BACK_N2B], xt2 + 128, NNn, 128, 16, 16, 128, 256, 0, 0);
  }

  // ---- 4. four attentions -> xall1/xall2 (N x 512) ----
  float* Qb = scr;
  float* Kb = Qb + NS * 128;
  float* Vb = Kb + NS * 128;
  // xall mid slice = xt
  k_copy_cols<<<EW(NS * 256), 256, 0, stream>>>(xt1, 256, xall1 + 128, 512, NNn, 256);
  k_copy_cols<<<EW(NS * 256), 256, 0, stream>>>(xt2, 256, xall2 + 128, 512, NNn, 256);
  // xst_w = attn(q=xt1, k=xs1, v=xs1) -> xall1[:,384:512]
  GEMM(xt1, P[P_ATT_W_QW], P[P_ATT_W_QB], Qb, NNn, 128, 256, 256, 128, 128, 0, 0);
  GEMM(xs1, P[P_ATT_W_KW], P[P_ATT_W_KB], Kb, NNn, 128, 128, 128, 128, 128, 0, 0);
  GEMM(xs1, P[P_ATT_W_VW], P[P_ATT_W_VB], Vb, NNn, 128, 128, 128, 128, 128, 0, 0);
  k_attn<<<NNn / 16, 32, 0, stream>>>(Qb, Kb, Vb, xall1 + 384, NNn, 512, 0);
  // xst_s = attn(q=xs2, k=xt2, v=xt2) -> xall2[:,384:512]
  GEMM(xs2, P[P_ATT_S_QW], P[P_ATT_S_QB], Qb, NNn, 128, 128, 128, 128, 128, 0, 0);
  GEMM(xt2, P[P_ATT_S_KW], P[P_ATT_S_KB], Kb, NNn, 128, 256, 256, 128, 128, 0, 0);
  GEMM(xt2, P[P_ATT_S_VW], P[P_ATT_S_VB], Vb, NNn, 128, 256, 256, 128, 128, 0, 0);
  k_attn<<<NNn / 16, 32, 0, stream>>>(Qb, Kb, Vb, xall2 + 384, NNn, 512, 0);
  // xup_w = lin(xs1,o) + attn(q=xs1, k=xt1, v=xt1) -> xall1[:,0:128]
  GEMM(xs1, P[P_ATT_WU_OW], P[P_ATT_WU_OB], xall1, NNn, 128, 128, 128, 128, 512, 0, 0);
  GEMM(xs1, P[P_ATT_WU_QW], P[P_ATT_WU_QB], Qb, NNn, 128, 128, 128, 128, 128, 0, 0);
  GEMM(xt1, P[P_ATT_WU_KW], P[P_ATT_WU_KB], Kb, NNn, 128, 256, 256, 128, 128, 0, 0);
  GEMM(xt1, P[P_ATT_WU_VW], P[P_ATT_WU_VB], Vb, NNn, 128, 256, 256, 128, 128, 0, 0);
  k_attn<<<NNn / 16, 32, 0, stream>>>(Qb, Kb, Vb, xall1, NNn, 512, 1);
  // xup_s = lin(xs2,o) + attn(q=xt2, k=xs2, v=xs2) -> xall2[:,0:128]
  GEMM(xs2, P[P_ATT_SU_OW], P[P_ATT_SU_OB], xall2, NNn, 128, 128, 128, 128, 512, 0, 0);
  GEMM(xt2, P[P_ATT_SU_QW], P[P_ATT_SU_QB], Qb, NNn, 128, 256, 256, 128, 128, 0, 0);
  GEMM(xs2, P[P_ATT_SU_KW], P[P_ATT_SU_KB], Kb, NNn, 128, 128, 128, 128, 128, 0, 0);
  GEMM(xs2, P[P_ATT_SU_VW], P[P_ATT_SU_VB], Vb, NNn, 128, 128, 128, 128, 128, 0, 0);
  k_attn<<<NNn / 16, 32, 0, stream>>>(Qb, Kb, Vb, xall2, NNn, 512, 1);

  // ---- 5. pooling masks + fc ----
  float* ctx = scr;                 // N x 512
  float* down = ctx + NS * 512;     // N x 512
  GEMM(xall2, P[P_FCC_W], P[P_FCC_B], ctx, NNn, 512, 512, 512, 512, 512, 0, 0);
  GEMM(xall2, P[P_FCD_W], P[P_FCD_B], down, NNn, 512, 512, 512, 512, 512, 0, 0);
  // x_out1 = [mask_in1 @ xall1 | dmask @ down]
  GEMM(mask_in1, xall1, nullptr, xout1, NNn, 512, NNn, NNn, 512, 1024, 0, 0);
  GEMM(dmask, down, nullptr, xout1 + 512, NNn, 512, NNn, NNn, 512, 1024, 0, 0);
  // xo3cat = [xall2 | cmask @ ctx]
  k_copy_cols<<<EW(NS * 512), 256, 0, stream>>>(xall2, 512, xo3cat, 1024, NNn, 512);
  GEMM(cmask, ctx, nullptr, xo3cat + 512, NNn, 512, NNn, NNn, 512, 1024, 0, 0);
  // xo3pack = mask_pack2 @ xo3cat ; xout3 = mask_in2 @ xo3cat
  GEMM(mask_pack2, xo3cat, nullptr, xo3pack, NNn, 1024, NNn, NNn, 1024, 1024, 0, 0);
  GEMM(mask_in2, xo3cat, nullptr, xout3, NNn, 1024, NNn, NNn, 1024, 1024, 0, 0);

  // ---- 6. heads ----
  float* hid = scr;  // up to N x 2048
  // Node1
  GEMM(xout1, P[P_OH_N1W], P[P_OH_N1B], hid, NNn, 1024, 1024, 1024, 1024, 1024, 0, 1);
  k_rowdot<<<NNn, 32, 0, stream>>>(hid, P[P_OUT_N1W], P[P_OUT_N1B], out, 1024);
  // pack: x_out2 = [x_out1 | xo3pack], split over K
  GEMM(xout1, P[P_OH_PKW], nullptr, hid, NNn, 2048, 1024, 1024, 2048, 2048, 0, 0);
  GEMM(xo3pack, P[P_OH_PKW] + (size_t)1024 * 2048, P[P_OH_PKB], hid, NNn, 2048,
       1024, 1024, 2048, 2048, 1, 1);
  k_rowdot<<<NNn, 32, 0, stream>>>(hid, P[P_OUT_PKW], P[P_OUT_PKB], out + NNn, 2048);
  // Node2
  GEMM(xout3, P[P_OH_N2W], P[P_OH_N2B], hid, NNn, 1024, 1024, 1024, 1024, 1024, 0, 1);
  k_rowdot<<<NNn, 32, 0, stream>>>(hid, P[P_OUT_N2W], P[P_OUT_N2B], out + 2 * NNn, 1024);
}